// AxialAttention_54752243089768
// MI455X (gfx1250) — compile-verified
//
#include <hip/hip_runtime.h>
#include <hip/hip_bf16.h>

typedef float v2f __attribute__((ext_vector_type(2)));
typedef float v8f __attribute__((ext_vector_type(8)));

namespace {

constexpr int NB = 516, NL = 129, NC = 512, ND = 2048;
constexpr int NH = 8, NDK = 64, NDV = 128;
constexpr int MPAD = 144, LPAD = 144;     // padded attention dims (multiple of 16)
constexpr int NBH = NB * NH;              // 4128
constexpr int BLROWS = NB * NL;           // 66564
constexpr float BN_EPS = 1e-3f;

// D = A(16x4) * B(4x16) + C, fp32 WMMA. Per-lane: A,B = 2 floats, C/D = 8 floats.
// A layout: lane(lo=lane&15,hi=lane>>4) holds A[M=lo][K=2*hi], A[M=lo][K=2*hi+1]
// B layout: lane holds B[K=2*hi][N=lo], B[K=2*hi+1][N=lo]
// C/D layout: VGPR r on lane -> element (M = r + 8*hi, N = lo)
__device__ __forceinline__ v8f wmma4(v2f a, v2f b, v8f c) {
  return __builtin_amdgcn_wmma_f32_16x16x4_f32(false, a, false, b, (short)0, c,
                                               false, false);
}

__device__ __forceinline__ void store_qkv(float y, int grow, int n,
                                          float* __restrict__ q,
                                          float* __restrict__ k,
                                          float* __restrict__ v) {
  const int bi = grow / NL, li = grow - bi * NL;
  if (n < NH * NDK) {                                // q block
    q[(((size_t)bi * NH + (n >> 6)) * NL + li) * NDK + (n & 63)] = y;
  } else if (n < 2 * NH * NDK) {                     // k block
    const int nn = n - NH * NDK;
    k[(((size_t)bi * NH + (nn >> 6)) * NL + li) * NDK + (nn & 63)] = y;
  } else {                                           // v block
    const int nn = n - 2 * NH * NDK;
    v[(((size_t)bi * NH + (nn >> 7)) * NL + li) * NDV + (nn & 127)] = y;
  }
}

// ---------------------------------------------------------------------------
// Kernel 1: qkv = X[BL,512] * Wk[512,2048], fused BN, scatter to q/k/v layouts
// Block = 128 threads = 4 waves; each wave owns a 32x32 output tile
// (4 accumulators -> 4 WMMA per 6 vmem ops in the k-loop).
// ---------------------------------------------------------------------------
__global__ void qkv_gemm_bn(const float* __restrict__ X,
                            const float* __restrict__ Wk,
                            const float* __restrict__ ga,
                            const float* __restrict__ be,
                            const float* __restrict__ mu,
                            const float* __restrict__ va,
                            float* __restrict__ q, float* __restrict__ k,
                            float* __restrict__ v) {
  const int lane = threadIdx.x & 31;
  const int wave = threadIdx.x >> 5;
  const int hi = lane >> 4, lo = lane & 15;
  const int row0 = blockIdx.y * 32;                 // over B*L
  const int col0 = (blockIdx.x * 4 + wave) * 32;    // over D=2048
  const int am0 = min(row0 + lo, BLROWS - 1);       // clamp padded rows
  const int am1 = min(row0 + 16 + lo, BLROWS - 1);
  const float* arow0 = X + (size_t)am0 * NC;
  const float* arow1 = X + (size_t)am1 * NC;
  const int n0 = col0 + lo;                         // lane's output columns
  const int n1 = col0 + 16 + lo;

  v8f acc00 = {0.f, 0.f, 0.f, 0.f, 0.f, 0.f, 0.f, 0.f};
  v8f acc01 = acc00, acc10 = acc00, acc11 = acc00;
  for (int kb = 0; kb < NC; kb += 4) {
    const int kr = kb + 2 * hi;
    __builtin_prefetch(arow0 + kb + 32, 0, 0);      // global_prefetch_b8
    v2f a0 = *(const v2f*)(arow0 + kr);
    v2f a1 = *(const v2f*)(arow1 + kr);
    v2f b0, b1;
    b0.x = Wk[(size_t)kr * ND + n0];
    b0.y = Wk[(size_t)(kr + 1) * ND + n0];
    b1.x = Wk[(size_t)kr * ND + n1];
    b1.y = Wk[(size_t)(kr + 1) * ND + n1];
    acc00 = wmma4(a0, b0, acc00);
    acc01 = wmma4(a0, b1, acc01);
    acc10 = wmma4(a1, b0, acc10);
    acc11 = wmma4(a1, b1, acc11);
  }

  const float sc0 = ga[n0] * rsqrtf(va[n0] + BN_EPS);
  const float sh0 = be[n0] - sc0 * mu[n0];
  const float sc1 = ga[n1] * rsqrtf(va[n1] + BN_EPS);
  const float sh1 = be[n1] - sc1 * mu[n1];
#pragma unroll
  for (int r = 0; r < 8; ++r) {
    const int g0 = row0 + r + 8 * hi;
    const int g1 = g0 + 16;
    if (g0 < BLROWS) {
      store_qkv(sc0 * acc00[r] + sh0, g0, n0, q, k, v);
      store_qkv(sc1 * acc01[r] + sh1, g0, n1, q, k, v);
    }
    if (g1 < BLROWS) {
      store_qkv(sc0 * acc10[r] + sh0, g1, n0, q, k, v);
      store_qkv(sc1 * acc11[r] + sh1, g1, n1, q, k, v);
    }
  }
}

// ---------------------------------------------------------------------------
// Kernel 2: logits[b,h,l,m] = a1*(q.k) + a2*(q.Tq[m-l+128]) + a3*(k.Tk[m-l+128]) + cb
// One wave per (bh, 16x16 tile). RPE terms via Toeplitz-band GEMMs Uq,Uk (16x32),
// staged in LDS, then gathered along diagonals.
// ---------------------------------------------------------------------------
__global__ void logits_kernel(const float* __restrict__ q,
                              const float* __restrict__ k,
                              const float* __restrict__ Tq,
                              const float* __restrict__ Tk,
                              const float* __restrict__ gs,
                              const float* __restrict__ bs,
                              const float* __restrict__ ms,
                              const float* __restrict__ vs,
                              float* __restrict__ Wbuf) {
  __shared__ float Uq_s[16][32];
  __shared__ float Uk_s[16][32];
  const int lane = threadIdx.x & 31;
  const int hi = lane >> 4, lo = lane & 15;
  const int bh = blockIdx.y;
  const int h = bh % NH;
  const int lt = blockIdx.x / 9, mt = blockIdx.x % 9;
  const int l0 = lt * 16, m0 = mt * 16;
  const float* qb = q + (size_t)bh * NL * NDK;
  const float* kb = k + (size_t)bh * NL * NDK;
  const int lA = min(l0 + lo, NL - 1);
  const int mB = min(m0 + lo, NL - 1);
  const int d0 = m0 - l0 + 113;                      // base diagonal index
  const int dqa = min(max(d0 + lo, 0), 256);         // table row, cols 0..15
  const int dqb = min(max(d0 + 16 + lo, 0), 256);    // table row, cols 16..31

  v8f accS = {0.f, 0.f, 0.f, 0.f, 0.f, 0.f, 0.f, 0.f};
  v8f aq0 = accS, aq1 = accS, ak0 = accS, ak1 = accS;
  for (int kk = 0; kk < NDK; kk += 4) {
    const int kr = kk + 2 * hi;
    // A-layout q row; k row serves as both A-layout (Uk) and B-layout (content)
    v2f aQ = *(const v2f*)(qb + (size_t)lA * NDK + kr);
    v2f aK = *(const v2f*)(kb + (size_t)mB * NDK + kr);
    accS = wmma4(aQ, aK, accS);                      // S = Q * K^T
    v2f bq0 = *(const v2f*)(Tq + (size_t)dqa * NDK + kr);
    v2f bq1 = *(const v2f*)(Tq + (size_t)dqb * NDK + kr);
    aq0 = wmma4(aQ, bq0, aq0);                       // Uq = Q * Tq[band]^T
    aq1 = wmma4(aQ, bq1, aq1);
    v2f bk0 = *(const v2f*)(Tk + (size_t)dqa * NDK + kr);
    v2f bk1 = *(const v2f*)(Tk + (size_t)dqb * NDK + kr);
    ak0 = wmma4(aK, bk0, ak0);                       // Uk = K * Tk[band]^T
    ak1 = wmma4(aK, bk1, ak1);
  }
#pragma unroll
  for (int r = 0; r < 8; ++r) {
    Uq_s[r + 8 * hi][lo] = aq0[r];
    Uq_s[r + 8 * hi][16 + lo] = aq1[r];
    Uk_s[r + 8 * hi][lo] = ak0[r];
    Uk_s[r + 8 * hi][16 + lo] = ak1[r];
  }
  __syncthreads();

  const float a1 = gs[h] * rsqrtf(vs[h] + BN_EPS);
  const float a2 = gs[NH + h] * rsqrtf(vs[NH + h] + BN_EPS);
  const float a3 = gs[2 * NH + h] * rsqrtf(vs[2 * NH + h] + BN_EPS);
  const float cb = (bs[h] - a1 * ms[h]) + (bs[NH + h] - a2 * ms[NH + h]) +
                   (bs[2 * NH + h] - a3 * ms[2 * NH + h]);
  float* wt = Wbuf + (size_t)bh * LPAD * MPAD;
#pragma unroll
  for (int r = 0; r < 8; ++r) {
    const int li = r + 8 * hi, mi = lo;
    const int l = l0 + li, m = m0 + mi;
    if (l < NL && m < NL) {
      const int c = mi - li + 15;                    // = (m-l+128) - d0, in [0,30]
      wt[(size_t)l * MPAD + m] =
          a1 * accS[r] + a2 * Uq_s[li][c] + a3 * Uk_s[mi][c] + cb;
    }
  }
}

// ---------------------------------------------------------------------------
// Kernel 3: row softmax over m (129 valid cols), zero-fill padding to 144.
// 4 waves per block, one wave per row.
// ---------------------------------------------------------------------------
__global__ void softmax_kernel(float* __restrict__ Wbuf) {
  const int lane = threadIdx.x & 31;
  const int wave = threadIdx.x >> 5;
  const long long row = (long long)blockIdx.x * 4 + wave;
  if (row >= (long long)NBH * NL) return;
  const int bh = (int)(row / NL);
  const int l = (int)(row - (long long)bh * NL);
  float* wr = Wbuf + ((size_t)bh * LPAD + l) * MPAD;

  float xv[5];
  float mx = -3.4e38f;
#pragma unroll
  for (int i = 0; i < 5; ++i) {
    const int m = lane + 32 * i;
    xv[i] = (m < NL) ? wr[m] : -3.4e38f;
    mx = fmaxf(mx, xv[i]);
  }
#pragma unroll
  for (int off = 16; off > 0; off >>= 1) mx = fmaxf(mx, __shfl_xor(mx, off, 32));
  float sum = 0.f;
#pragma unroll
  for (int i = 0; i < 5; ++i) {
    const int m = lane + 32 * i;
    const float e = (m < NL) ? __expf(xv[i] - mx) : 0.f;
    xv[i] = e;
    sum += e;
  }
#pragma unroll
  for (int off = 16; off > 0; off >>= 1) sum += __shfl_xor(sum, off, 32);
  const float inv = 1.f / sum;
#pragma unroll
  for (int i = 0; i < 5; ++i) {
    const int m = lane + 32 * i;
    if (m < MPAD) wr[m] = xv[i] * inv;               // cols 129..143 become 0
  }
}

// ---------------------------------------------------------------------------
// Kernel 4: out = BN(W*V) + BN(W' * Tv[band]) with W' the diagonal-skewed
// weights built in LDS. One wave per (bh, 16-l x 16-d tile).
// ---------------------------------------------------------------------------
__global__ void retrieve_out(const float* __restrict__ Wbuf,
                             const float* __restrict__ v,
                             const float* __restrict__ Tv,
                             const float* __restrict__ go,
                             const float* __restrict__ bo,
                             const float* __restrict__ mo,
                             const float* __restrict__ vo,
                             float* __restrict__ out) {
  __shared__ float wsk[16][MPAD];                    // skewed weights, 9 KB
  const int lane = threadIdx.x & 31;
  const int hi = lane >> 4, lo = lane & 15;
  const int bh = blockIdx.y;
  const int b = bh / NH, h = bh - b * NH;
  const int lt = blockIdx.x >> 3, dt = blockIdx.x & 7;
  const int l0 = lt * 16, dc0 = dt * 16;
  const float* wb = Wbuf + (size_t)bh * LPAD * MPAD;
  const float* vb = v + (size_t)bh * NL * NDV;

  // W'[li][t] = w[l0+li][t + li - 15]  (zeros outside [0,NL))
  for (int idx = lane; idx < 16 * MPAD; idx += 32) {
    const int li = idx / MPAD, t = idx - li * MPAD;
    const int m = t + li - 15;
    wsk[li][t] = (m >= 0 && m < NL) ? wb[(size_t)(l0 + li) * MPAD + m] : 0.f;
  }
  __syncthreads();

  const int dmin = 113 - l0;
  v8f acc1 = {0.f, 0.f, 0.f, 0.f, 0.f, 0.f, 0.f, 0.f};
  v8f acc2 = acc1;
  for (int kk = 0; kk < MPAD; kk += 4) {
    const int kr = kk + 2 * hi;
    // content retrieval: W[l, m] * v[m, d]   (padded w cols are 0)
    v2f a1v = *(const v2f*)(wb + (size_t)(l0 + lo) * MPAD + kr);
    const int mv0 = min(kr, NL - 1), mv1 = min(kr + 1, NL - 1);
    v2f b1v;
    b1v.x = vb[(size_t)mv0 * NDV + dc0 + lo];
    b1v.y = vb[(size_t)mv1 * NDV + dc0 + lo];
    acc1 = wmma4(a1v, b1v, acc1);
    // value-RPE retrieval: W'[l, t] * Tv[t + dmin, d]
    v2f a2v = *(const v2f*)&wsk[lo][kr];
    const int r0 = min(max(kr + dmin, 0), 256);
    const int r1 = min(max(kr + 1 + dmin, 0), 256);
    v2f b2v;
    b2v.x = Tv[(size_t)r0 * NDV + dc0 + lo];
    b2v.y = Tv[(size_t)r1 * NDV + dc0 + lo];
    acc2 = wmma4(a2v, b2v, acc2);
  }

  const int ix = h * NDV + dc0 + lo;
  const float s1 = go[ix] * rsqrtf(vo[ix] + BN_EPS);
  const float s2 = go[NH * NDV + ix] * rsqrtf(vo[NH * NDV + ix] + BN_EPS);
  const float cc = (bo[ix] - s1 * mo[ix]) +
                   (bo[NH * NDV + ix] - s2 * mo[NH * NDV + ix]);
#pragma unroll
  for (int r = 0; r < 8; ++r) {
    const int l = l0 + r + 8 * hi;
    if (l < NL)
      out[((size_t)b * NL + l) * (NH * NDV) + ix] =
          s1 * acc1[r] + s2 * acc2[r] + cc;
  }
}

}  // namespace

extern "C" void kernel_launch(void* const* d_in, const int* in_sizes, int n_in,
                              void* d_out, int out_size, void* d_ws,
                              size_t ws_size, hipStream_t stream) {
  const float* x   = (const float*)d_in[0];
  const float* wk  = (const float*)d_in[1];
  const float* gq  = (const float*)d_in[2];
  const float* bq  = (const float*)d_in[3];
  const float* mq  = (const float*)d_in[4];
  const float* vq  = (const float*)d_in[5];
  const float* tq  = (const float*)d_in[6];
  const float* tk  = (const float*)d_in[7];
  const float* tv  = (const float*)d_in[8];
  const float* gsim = (const float*)d_in[9];
  const float* bsim = (const float*)d_in[10];
  const float* msim = (const float*)d_in[11];
  const float* vsim = (const float*)d_in[12];
  const float* gout = (const float*)d_in[13];
  const float* bout = (const float*)d_in[14];
  const float* mout = (const float*)d_in[15];
  const float* vout = (const float*)d_in[16];
  float* out = (float*)d_out;
  (void)in_sizes; (void)n_in; (void)out_size; (void)ws_size;

  float* ws = (float*)d_ws;
  const size_t QKSZ = (size_t)NBH * NL * NDK;   // 34,080,768 floats
  const size_t VSZ  = (size_t)NBH * NL * NDV;   // 68,161,536 floats
  float* qp = ws;
  float* kp = ws + QKSZ;
  float* vp = ws + 2 * QKSZ;
  float* Wp = vp + VSZ;                         // NBH * 144 * 144 floats

  // 1) QKV projection + BN (32x32 tile per wave, 4 waves/block -> 128 cols)
  qkv_gemm_bn<<<dim3(ND / 128, (BLROWS + 31) / 32), 128, 0, stream>>>(
      x, wk, gq, bq, mq, vq, qp, kp, vp);
  // 2) attention logits (content + both RPE sims, BN folded)
  logits_kernel<<<dim3(9 * 9, NBH), 32, 0, stream>>>(qp, kp, tq, tk, gsim,
                                                     bsim, msim, vsim, Wp);
  // 3) softmax over keys
  const long long rows = (long long)NBH * NL;
  softmax_kernel<<<dim3((unsigned)((rows + 3) / 4)), 128, 0, stream>>>(Wp);
  // 4) retrievals + output BN
  retrieve_out<<<dim3(9 * 8, NBH), 32, 0, stream>>>(Wp, vp, tv, gout, bout,
                                                    mout, vout, out);
}